// PatchMatch_58909771432325
// MI455X (gfx1250) — compile-verified
//
#include <hip/hip_runtime.h>
#include <math.h>

// ---------------------------------------------------------------------------
// PatchMatch (b=2, c=32, 256x256, patch 3, 5 iters, radius 4) for gfx1250.
//
// Key algebraic reduction: both source and target patches are gathered at the
// SAME nnf coordinate, so the SSD is a pure function D(y,x) of the coordinate.
//   D(y,x) = 3x3 zero-padded box-sum of E, E[y,x] = sum_c (src-tgt)^2.
// => propagate needs NO distance evals (neighbor's cur_d is already known),
//    random search needs 4 table lookups/pixel/iter, rows are independent.
//
// Roofline: E precompute reads 2*2*32*256*256*4B = 33.5MB -> ~1.4us @23.3TB/s
// (bandwidth bound). D table = 512KB, L2 resident. Channel reduction runs on
// the matrix pipe via V_WMMA_F32_16X16X4_F32 with a ones A-matrix (exact f32).
// ---------------------------------------------------------------------------

typedef __attribute__((ext_vector_type(2))) float v2f;
typedef __attribute__((ext_vector_type(8))) float v8f;

#define BB     2
#define CC     32
#define HH     256
#define WW     256
#define HW     (HH * WW)
#define ITERS  5
#define RADIUS 4

// ---------------------------------------------------------------------------
// Kernel 1a: E[b,p] = sum_c (src-tgt)^2, channel reduction on the WMMA pipe.
// One wave handles 16 pixels. B operand (4x16 f32, 2 VGPRs):
//   VGPR0: lanes0-15 -> K=0 row, lanes16-31 -> K=2 row; VGPR1: K=1 / K=3.
// A = ones(16x4) => C[m,n] = sum_k B[k,n]; any K/M permutation is harmless.
// ---------------------------------------------------------------------------
__global__ __launch_bounds__(256) void pm_e_wmma(const float* __restrict__ src,
                                                 const float* __restrict__ tgt,
                                                 float* __restrict__ E) {
  const int lane = threadIdx.x & 31;
  const int wave = threadIdx.x >> 5;
  const int waveGlobal = blockIdx.x * 8 + wave;       // 8192 waves total
  const int pixBase = waveGlobal * 16;                // 16 pixels / wave
  const int half = lane >> 4;
  const int pix  = pixBase + (lane & 15);             // linear over b*H*W
  const int img  = pix >> 16;                         // / (H*W)
  const int p    = pix & 0xFFFF;
  const long long base = (long long)img * CC * HW + p;

  v2f a; a.x = 1.0f; a.y = 1.0f;                      // ones A-matrix
  v8f acc = {};

#pragma unroll
  for (int k = 0; k < 8; ++k) {                       // 8 chunks of 4 channels
    const int cA = 4 * k + 2 * half;
    const long long o0 = base + (long long)cA * HW;
    const long long o1 = o0 + HW;
    const float d0 = src[o0] - tgt[o0];
    const float d1 = src[o1] - tgt[o1];
    v2f bm; bm.x = d0 * d0; bm.y = d1 * d1;
    acc = __builtin_amdgcn_wmma_f32_16x16x4_f32(
        /*neg_a=*/false, a, /*neg_b=*/false, bm,
        /*c_mod=*/(short)0, acc, /*reuse_a=*/false, /*reuse_b=*/false);
  }
  // Every row of C holds the column sums; lanes 16-31 duplicate lanes 0-15.
  if (half == 0) E[pix] = acc[0];
}

// ---------------------------------------------------------------------------
// Kernel 1b: D = 3x3 zero-boundary box sum of E. Trivial, L2-resident.
// ---------------------------------------------------------------------------
__global__ __launch_bounds__(256) void pm_box3x3(const float* __restrict__ E,
                                                 float* __restrict__ D) {
  const int x = threadIdx.x;
  const int y = blockIdx.x & 255;
  const int img = blockIdx.x >> 8;
  const float* Eb = E + img * HW;
  float s = 0.f;
#pragma unroll
  for (int dy = -1; dy <= 1; ++dy) {
    const int yy = y + dy;
    if (yy < 0 || yy >= HH) continue;
#pragma unroll
    for (int dx = -1; dx <= 1; ++dx) {
      const int xx = x + dx;
      if (xx < 0 || xx >= WW) continue;
      s += Eb[yy * WW + xx];
    }
  }
  D[img * HW + y * WW + x] = s;
}

// ---------------------------------------------------------------------------
// Deterministic counter-based RNG (splitmix64) + Box-Muller normals.
// ---------------------------------------------------------------------------
__device__ __forceinline__ unsigned long long splitmix64(unsigned long long z) {
  z += 0x9E3779B97F4A7C15ull;
  z = (z ^ (z >> 30)) * 0xBF58476D1CE4E5B9ull;
  z = (z ^ (z >> 27)) * 0x94D049BB133111EBull;
  return z ^ (z >> 31);
}

__device__ __forceinline__ float lookupD(const float* __restrict__ Db,
                                         float cy, float cx) {
  const int yi = (int)rintf(fminf(fmaxf(cy, 0.f), 255.f));
  const int xi = (int)rintf(fminf(fmaxf(cx, 0.f), 255.f));
  return Db[yi * WW + xi];
}

// ---------------------------------------------------------------------------
// Kernel 2: full 5-iteration PatchMatch loop. One workgroup per (batch,row);
// rows are independent because roll() is along W only and distances come from
// the precomputed D table. Propagation = LDS exchange of (y,x,d).
// ---------------------------------------------------------------------------
__global__ __launch_bounds__(256) void pm_iterate(const float* __restrict__ D,
                                                  float* __restrict__ out) {
  const int j   = threadIdx.x;
  const int y   = blockIdx.x & 255;
  const int img = blockIdx.x >> 8;
  const int idx = (img * HH + y) * WW + j;            // global pixel id
  const float* Db = D + img * HW;

  __shared__ float sy[WW], sx[WW], sd[WW];

  // init nnf ~ U[0,255]^2 (deterministic counter hash)
  unsigned long long h = splitmix64(0x243F6A8885A308D3ull ^ (unsigned long long)idx);
  float ny = (float)((unsigned)(h >> 40)) * 0x1p-24f * 255.0f;
  float nx = (float)(((unsigned)h) >> 8) * 0x1p-24f * 255.0f;
  float cur_d = lookupD(Db, ny, nx);

#pragma unroll
  for (int t = 0; t < ITERS; ++t) {
    // ---- propagate direction +1: candidate = left neighbor (wrap) ----
    sy[j] = ny; sx[j] = nx; sd[j] = cur_d;
    __syncthreads();
    {
      const int l = (j + WW - 1) & (WW - 1);
      const float dy = sy[l], dx = sx[l], dd = sd[l];
      __syncthreads();
      if (dd < cur_d) { ny = dy; nx = dx; cur_d = dd; }
    }
    // ---- propagate direction -1: candidate = right neighbor (wrap) ----
    sy[j] = ny; sx[j] = nx; sd[j] = cur_d;
    __syncthreads();
    {
      const int r = (j + 1) & (WW - 1);
      const float dy = sy[r], dx = sx[r], dd = sd[r];
      __syncthreads();
      if (dd < cur_d) { ny = dy; nx = dx; cur_d = dd; }
    }
    // ---- random search: 4 shrinking Gaussian trials ----
#pragma unroll
    for (int i = 0; i < RADIUS; ++i) {
      const unsigned long long ctr =
          ((unsigned long long)(1 + t * RADIUS + i) << 40) ^
          (unsigned long long)idx;
      const unsigned long long hh = splitmix64(ctr ^ 0x452821E638D01377ull);
      float u0 = (float)((unsigned)(hh >> 40)) * 0x1p-24f;
      const float u1 = (float)(((unsigned)hh) >> 8) * 0x1p-24f;
      u0 = fmaxf(u0, 1.0e-12f);                        // avoid log(0)
      const float rr = sqrtf(-2.0f * __logf(u0));
      const float ang = 6.28318530718f * u1;
      const float n0 = rr * __cosf(ang);
      const float n1 = rr * __sinf(ang);
      const float scale = 1.0f / (float)(1 << i);      // 2^-i
      const float cy = fminf(fmaxf(ny + n0 * scale, 0.f), 255.f);
      const float cx = fminf(fmaxf(nx + n1 * scale, 0.f), 255.f);
      const float rd = lookupD(Db, cy, cx);
      if (rd < cur_d) { ny = cy; nx = cx; cur_d = rd; }
    }
  }

  // out layout: [b, 2, h, w]
  out[((img * 2 + 0) * HH + y) * WW + j] = ny;
  out[((img * 2 + 1) * HH + y) * WW + j] = nx;
}

// ---------------------------------------------------------------------------
extern "C" void kernel_launch(void* const* d_in, const int* in_sizes, int n_in,
                              void* d_out, int out_size, void* d_ws, size_t ws_size,
                              hipStream_t stream) {
  const float* src = (const float*)d_in[0];   // [2,32,256,256] f32
  const float* tgt = (const float*)d_in[1];   // [2,32,256,256] f32
  float* out = (float*)d_out;                 // [2,2,256,256] f32

  // workspace: E (512KB) then D (512KB)
  float* E = (float*)d_ws;
  float* D = E + BB * HW;

  pm_e_wmma<<<dim3(1024), dim3(256), 0, stream>>>(src, tgt, E);
  pm_box3x3<<<dim3(BB * HH), dim3(256), 0, stream>>>(E, D);
  pm_iterate<<<dim3(BB * HH), dim3(256), 0, stream>>>(D, out);
}